// BotGAT_5901285065194
// MI455X (gfx1250) — compile-verified
//
#include <hip/hip_runtime.h>
#include <hip/hip_bf16.h>
#include <math.h>

#define NNODES 100000
#define NEDGES 1600000
#define HID    128

typedef __attribute__((ext_vector_type(2))) float v2f;
typedef __attribute__((ext_vector_type(8))) float v8f;

// ---------------------------------------------------------------------------
// Fast fp32 WMMA GEMM (compile-time K, K%4==0, lda/ldw even => 8B-aligned rows):
//   C[:, ccol0 .. ccol0+NT*16) = act(A[nrows,KV] @ W[NT*16,KV]^T + bias)
// One wave = 16-row stripe x (NT*16) cols.  A fragment reused across NT tiles.
// A-frag (ISA 16x4 f32):  lane = kh*16 + m, v2f = A[m][k + 2kh + {0,1}]
// B-frag (4x16, B=W^T):   lane = kh*16 + n, v2f = W[n][k + 2kh + {0,1}]
// C/D: lane = half*16 + n, elem v -> row (v + 8*half), col n
// ---------------------------------------------------------------------------
template<int NT, int KV>
__global__ __launch_bounds__(256) void gemm_wmma_f32_fast(
    const float* __restrict__ A, int lda,
    const float* __restrict__ W, int ldw,
    const float* __restrict__ bias,
    float* __restrict__ C, int ldc, int ccol0,
    int nrows, int do_lrelu, float slope)
{
    const int wave = blockIdx.x * (blockDim.x >> 5) + (threadIdx.x >> 5);
    const int row0 = wave << 4;
    if (row0 >= nrows) return;

    const int lane = threadIdx.x & 31;
    const int m    = lane & 15;   // A row within tile / B col within tile
    const int kh   = lane >> 4;   // K-half selector

    v8f zero = {};
    v8f acc[NT];
#pragma unroll
    for (int t = 0; t < NT; ++t) acc[t] = zero;

    // v2f-typed row pointers (8B aligned: lda/ldw even)
    const v2f* __restrict__ arow = (const v2f*)(A + (size_t)(row0 + m) * lda);
    const v2f* __restrict__ wrow[NT];
#pragma unroll
    for (int t = 0; t < NT; ++t)
        wrow[t] = (const v2f*)(W + (size_t)(t * 16 + m) * ldw);

    __builtin_prefetch((const float*)arow, 0, 0);   // global_prefetch_b8

    // k2 indexes v2f units; fragment index = k2 + kh
#pragma unroll 4
    for (int k2 = 0; k2 < KV / 2; k2 += 2) {
        const v2f a = arow[k2 + kh];
#pragma unroll
        for (int t = 0; t < NT; ++t) {
            const v2f b = wrow[t][k2 + kh];
            acc[t] = __builtin_amdgcn_wmma_f32_16x16x4_f32(
                false, a, false, b, (short)0, acc[t], false, false);
        }
    }

#pragma unroll
    for (int t = 0; t < NT; ++t) {
        const int wc = t * 16 + m;          // weight-local output index
        const int cc = ccol0 + wc;          // global output column
        const float bv = bias ? bias[wc] : 0.0f;
#pragma unroll
        for (int v = 0; v < 8; ++v) {
            const int r = row0 + v + 8 * kh;
            float val = acc[t][v] + bv;
            if (do_lrelu) val = (val > 0.0f) ? val : val * slope;
            C[(size_t)r * ldc + cc] = val;
        }
    }
}

// ---------------------------------------------------------------------------
// Guarded GEMM for tiny ragged K (5, 3): scalar predicated loads, 2 k-steps.
// ---------------------------------------------------------------------------
template<int NT>
__global__ __launch_bounds__(256) void gemm_wmma_f32_small(
    const float* __restrict__ A, int lda,
    const float* __restrict__ W, int ldw,
    const float* __restrict__ bias,
    float* __restrict__ C, int ldc, int ccol0,
    int nrows, int K, int do_lrelu, float slope)
{
    const int wave = blockIdx.x * (blockDim.x >> 5) + (threadIdx.x >> 5);
    const int row0 = wave << 4;
    if (row0 >= nrows) return;

    const int lane = threadIdx.x & 31;
    const int m    = lane & 15;
    const int kh   = lane >> 4;

    v8f zero = {};
    v8f acc[NT];
#pragma unroll
    for (int t = 0; t < NT; ++t) acc[t] = zero;

    const float* __restrict__ arow = A + (size_t)(row0 + m) * lda;

    for (int k = 0; k < K; k += 4) {
        const int kk = k + 2 * kh;
        v2f a;
        a.x = (kk     < K) ? arow[kk]     : 0.0f;
        a.y = (kk + 1 < K) ? arow[kk + 1] : 0.0f;
#pragma unroll
        for (int t = 0; t < NT; ++t) {
            const float* __restrict__ wr = W + (size_t)(t * 16 + m) * ldw;
            v2f b;
            b.x = (kk     < K) ? wr[kk]     : 0.0f;
            b.y = (kk + 1 < K) ? wr[kk + 1] : 0.0f;
            acc[t] = __builtin_amdgcn_wmma_f32_16x16x4_f32(
                false, a, false, b, (short)0, acc[t], false, false);
        }
    }

#pragma unroll
    for (int t = 0; t < NT; ++t) {
        const int wc = t * 16 + m;
        const int cc = ccol0 + wc;
        const float bv = bias ? bias[wc] : 0.0f;
#pragma unroll
        for (int v = 0; v < 8; ++v) {
            const int r = row0 + v + 8 * kh;
            float val = acc[t][v] + bv;
            if (do_lrelu) val = (val > 0.0f) ? val : val * slope;
            C[(size_t)r * ldc + cc] = val;
        }
    }
}

// ---------------------------------------------------------------------------
// Per-(node,head) attention logits:  esrc = <xw_h, a_src_h>, edst = <xw_h, a_dst_h>
// ---------------------------------------------------------------------------
__global__ void attn_scores(const float* __restrict__ xw,
                            const float* __restrict__ a_s,
                            const float* __restrict__ a_d,
                            float* __restrict__ esrc,
                            float* __restrict__ edst,
                            int n, int H, int D)
{
    int i = blockIdx.x * blockDim.x + threadIdx.x;   // node*H + h
    if (i >= n * H) return;
    const int node = i / H, h = i % H;
    const float4* __restrict__ row = (const float4*)(xw + (size_t)node * H * D + (size_t)h * D);
    const float4* __restrict__ as4 = (const float4*)(a_s + (size_t)h * D);
    const float4* __restrict__ ad4 = (const float4*)(a_d + (size_t)h * D);
    float ss = 0.0f, sd = 0.0f;
    for (int c = 0; c < D / 4; ++c) {
        float4 v = row[c], w0 = as4[c], w1 = ad4[c];
        ss += v.x * w0.x + v.y * w0.y + v.z * w0.z + v.w * w0.w;
        sd += v.x * w1.x + v.y * w1.y + v.z * w1.z + v.w * w1.w;
    }
    esrc[i] = ss;
    edst[i] = sd;
}

// ordered-uint encoding of float for atomicMax-based segment max
__device__ __forceinline__ unsigned fkey(float f) {
    unsigned u = __float_as_uint(f);
    return (u & 0x80000000u) ? ~u : (u | 0x80000000u);
}
__device__ __forceinline__ float fdec(unsigned k) {
    return (k & 0x80000000u) ? __uint_as_float(k & 0x7FFFFFFFu)
                             : __uint_as_float(~k);
}

__device__ __forceinline__ void edge_sd(const int* __restrict__ src,
                                        const int* __restrict__ dst,
                                        int i, int E, int& s, int& d)
{
    if (i < E) { s = src[i]; d = dst[i]; }
    else       { s = i - E;  d = i - E; }   // implicit self-loop
}

// pass 1: segment max of leaky_relu(esrc[s]+edst[d], 0.2)
__global__ void edge_max_k(const int* __restrict__ src, const int* __restrict__ dst,
                           int E, int n, int H,
                           const float* __restrict__ esrc, const float* __restrict__ edst,
                           unsigned* __restrict__ mkey)
{
    int i = blockIdx.x * blockDim.x + threadIdx.x;
    if (i >= E + n) return;
    int s, d; edge_sd(src, dst, i, E, s, d);
    for (int h = 0; h < H; ++h) {
        float e = esrc[s * H + h] + edst[d * H + h];
        e = (e > 0.0f) ? e : 0.2f * e;
        atomicMax(&mkey[d * H + h], fkey(e));
    }
}

// pass 2: segment sum of exp(e - max)
__global__ void edge_expsum_k(const int* __restrict__ src, const int* __restrict__ dst,
                              int E, int n, int H,
                              const float* __restrict__ esrc, const float* __restrict__ edst,
                              const unsigned* __restrict__ mkey,
                              float* __restrict__ den)
{
    int i = blockIdx.x * blockDim.x + threadIdx.x;
    if (i >= E + n) return;
    int s, d; edge_sd(src, dst, i, E, s, d);
    for (int h = 0; h < H; ++h) {
        float e = esrc[s * H + h] + edst[d * H + h];
        e = (e > 0.0f) ? e : 0.2f * e;
        float ex = __expf(e - fdec(mkey[d * H + h]));
        atomicAdd(&den[d * H + h], ex);
    }
}

// pass 3: out[d] += alpha * xw[s]; 32 threads/edge, float4 chunk each (HD==128)
__global__ void edge_accum_k(const int* __restrict__ src, const int* __restrict__ dst,
                             int E, int n, int H, int D,
                             const float* __restrict__ esrc, const float* __restrict__ edst,
                             const unsigned* __restrict__ mkey,
                             const float* __restrict__ den,
                             const float* __restrict__ xw,
                             float* __restrict__ out)
{
    const int HD = H * D;
    const int chunks = HD / 4;
    long long tid = (long long)blockIdx.x * blockDim.x + threadIdx.x;
    long long total = (long long)(E + n) * chunks;
    if (tid >= total) return;
    int i  = (int)(tid / chunks);
    int c0 = (int)(tid % chunks) * 4;
    int h  = c0 / D;
    int s, d; edge_sd(src, dst, i, E, s, d);

    float e = esrc[s * H + h] + edst[d * H + h];
    e = (e > 0.0f) ? e : 0.2f * e;
    float alpha = __expf(e - fdec(mkey[d * H + h])) / den[d * H + h];

    float4 v = *(const float4*)(xw + (size_t)s * HD + c0);
    float* o = out + (size_t)d * HD + c0;
    atomicAdd(o + 0, v.x * alpha);
    atomicAdd(o + 1, v.y * alpha);
    atomicAdd(o + 2, v.z * alpha);
    atomicAdd(o + 3, v.w * alpha);
}

// init accumulator with broadcast bias:  out[n][c] = b[c]   (fuses bias add)
__global__ void init_bias_k(float* __restrict__ x, const float* __restrict__ b, int n, int w)
{
    int i = blockIdx.x * blockDim.x + threadIdx.x;
    if (i >= n * w) return;
    x[i] = b[i % w];
}

__global__ void fill_f32_k(float* p, float v, int n)
{
    int i = blockIdx.x * blockDim.x + threadIdx.x;
    if (i < n) p[i] = v;
}
__global__ void fill_u32_k(unsigned* p, unsigned v, int n)
{
    int i = blockIdx.x * blockDim.x + threadIdx.x;
    if (i < n) p[i] = v;
}

// final 128 -> 2 projection (too narrow for a 16-wide tile; VALU dot)
__global__ void final_proj_k(const float* __restrict__ x,
                             const float* __restrict__ W,   // [2,128]
                             const float* __restrict__ b,   // [2]
                             float* __restrict__ out, int n)
{
    int i = blockIdx.x * blockDim.x + threadIdx.x;
    if (i >= n) return;
    const float4* r  = (const float4*)(x + (size_t)i * HID);
    const float4* w0 = (const float4*)(W);
    const float4* w1 = (const float4*)(W + HID);
    float a0 = 0.0f, a1 = 0.0f;
    for (int c = 0; c < HID / 4; ++c) {
        float4 v = r[c], u0 = w0[c], u1 = w1[c];
        a0 += v.x * u0.x + v.y * u0.y + v.z * u0.z + v.w * u0.w;
        a1 += v.x * u1.x + v.y * u1.y + v.z * u1.z + v.w * u1.w;
    }
    out[2 * i + 0] = a0 + b[0];
    out[2 * i + 1] = a1 + b[1];
}

// ---------------------------------------------------------------------------
static inline int cdiv(long long a, long long b) { return (int)((a + b - 1) / b); }

extern "C" void kernel_launch(void* const* d_in, const int* in_sizes, int n_in,
                              void* d_out, int out_size, void* d_ws, size_t ws_size,
                              hipStream_t stream)
{
    const int N = NNODES, E = NEDGES;

    const float* des  = (const float*)d_in[0];
    const float* twt  = (const float*)d_in[1];
    const float* nump = (const float*)d_in[2];
    const float* catp = (const float*)d_in[3];
    const int*   ei   = (const int*)  d_in[4];
    const float* Wd   = (const float*)d_in[5];
    const float* bd   = (const float*)d_in[6];
    const float* Wt   = (const float*)d_in[7];
    const float* bt   = (const float*)d_in[8];
    const float* Wn   = (const float*)d_in[9];
    const float* bn   = (const float*)d_in[10];
    const float* Wc   = (const float*)d_in[11];
    const float* bc   = (const float*)d_in[12];
    const float* Wi   = (const float*)d_in[13];
    const float* bi   = (const float*)d_in[14];
    const float* g1W  = (const float*)d_in[15];
    const float* g1as = (const float*)d_in[16];
    const float* g1ad = (const float*)d_in[17];
    const float* g1b  = (const float*)d_in[18];
    const float* g2W  = (const float*)d_in[19];
    const float* g2as = (const float*)d_in[20];
    const float* g2ad = (const float*)d_in[21];
    const float* g2b  = (const float*)d_in[22];
    const float* Wo1  = (const float*)d_in[23];
    const float* bo1  = (const float*)d_in[24];
    const float* Wo2  = (const float*)d_in[25];
    const float* bo2  = (const float*)d_in[26];

    const int* src = ei;
    const int* dst = ei + E;

    // workspace carve-up
    float*    buf0 = (float*)d_ws;                      // [N,128]
    float*    buf1 = buf0 + (size_t)N * HID;            // [N,128]
    float*    buf2 = buf1 + (size_t)N * HID;            // [N,128]
    float*    esrc = buf2 + (size_t)N * HID;            // [N,4]
    float*    edst = esrc + (size_t)N * 4;              // [N,4]
    unsigned* mkey = (unsigned*)(edst + (size_t)N * 4); // [N,4]
    float*    den  = (float*)(mkey + (size_t)N * 4);    // [N,4]

    const int stripes    = N / 16;                 // 6250 (N % 16 == 0)
    const int gemm_grid  = cdiv(stripes, 8);       // 8 waves / 256-thread block
    const int ET         = E + N;
    const unsigned NEGKEY = 0x00000000u;           // below fkey(-FLT_MAX)

    // ---- feature MLPs -> buf0 = concat(lrelu(...)) -------------------------
    gemm_wmma_f32_fast<2, 768><<<gemm_grid, 256, 0, stream>>>(des, 768, Wd, 768, bd, buf0, HID,  0, N, 1, 0.01f);
    gemm_wmma_f32_fast<2, 768><<<gemm_grid, 256, 0, stream>>>(twt, 768, Wt, 768, bt, buf0, HID, 32, N, 1, 0.01f);
    gemm_wmma_f32_small<2>    <<<gemm_grid, 256, 0, stream>>>(nump,  5, Wn,   5, bn, buf0, HID, 64, N, 5, 1, 0.01f);
    gemm_wmma_f32_small<2>    <<<gemm_grid, 256, 0, stream>>>(catp,  3, Wc,   3, bc, buf0, HID, 96, N, 3, 1, 0.01f);

    // ---- x = lrelu(h @ Wi^T + bi) -> buf1 ---------------------------------
    gemm_wmma_f32_fast<8, 128><<<gemm_grid, 256, 0, stream>>>(buf0, HID, Wi, HID, bi, buf1, HID, 0, N, 1, 0.01f);

    // ======================= GAT layer 1 (H=4, D=32) =======================
    {
        const int H = 4, D = 32;
        // xw = x @ g1_W^T -> buf2
        gemm_wmma_f32_fast<8, 128><<<gemm_grid, 256, 0, stream>>>(buf1, HID, g1W, HID, nullptr, buf2, HID, 0, N, 0, 0.0f);
        attn_scores<<<cdiv((long long)N * H, 256), 256, 0, stream>>>(buf2, g1as, g1ad, esrc, edst, N, H, D);
        fill_u32_k<<<cdiv((long long)N * H, 256), 256, 0, stream>>>(mkey, NEGKEY, N * H);
        fill_f32_k<<<cdiv((long long)N * H, 256), 256, 0, stream>>>(den, 0.0f, N * H);
        init_bias_k<<<cdiv((long long)N * HID, 256), 256, 0, stream>>>(buf0, g1b, N, HID);
        edge_max_k   <<<cdiv(ET, 256), 256, 0, stream>>>(src, dst, E, N, H, esrc, edst, mkey);
        edge_expsum_k<<<cdiv(ET, 256), 256, 0, stream>>>(src, dst, E, N, H, esrc, edst, mkey, den);
        edge_accum_k <<<cdiv((long long)ET * (HID / 4), 256), 256, 0, stream>>>(
            src, dst, E, N, H, D, esrc, edst, mkey, den, buf2, buf0);
    }

    // ======================= GAT layer 2 (H=1, D=128) ======================
    {
        const int H = 1, D = 128;
        gemm_wmma_f32_fast<8, 128><<<gemm_grid, 256, 0, stream>>>(buf0, HID, g2W, HID, nullptr, buf1, HID, 0, N, 0, 0.0f);
        attn_scores<<<cdiv((long long)N * H, 256), 256, 0, stream>>>(buf1, g2as, g2ad, esrc, edst, N, H, D);
        fill_u32_k<<<cdiv((long long)N * H, 256), 256, 0, stream>>>(mkey, NEGKEY, N * H);
        fill_f32_k<<<cdiv((long long)N * H, 256), 256, 0, stream>>>(den, 0.0f, N * H);
        init_bias_k<<<cdiv((long long)N * HID, 256), 256, 0, stream>>>(buf2, g2b, N, HID);
        edge_max_k   <<<cdiv(ET, 256), 256, 0, stream>>>(src, dst, E, N, H, esrc, edst, mkey);
        edge_expsum_k<<<cdiv(ET, 256), 256, 0, stream>>>(src, dst, E, N, H, esrc, edst, mkey, den);
        edge_accum_k <<<cdiv((long long)ET * (HID / 4), 256), 256, 0, stream>>>(
            src, dst, E, N, H, D, esrc, edst, mkey, den, buf1, buf2);
    }

    // ---- x = lrelu(x @ Wo1^T + bo1) -> buf1 -------------------------------
    gemm_wmma_f32_fast<8, 128><<<gemm_grid, 256, 0, stream>>>(buf2, HID, Wo1, HID, bo1, buf1, HID, 0, N, 1, 0.01f);

    // ---- out = x @ Wo2^T + bo2 -> d_out [N,2] -----------------------------
    final_proj_k<<<cdiv(N, 256), 256, 0, stream>>>(buf1, Wo2, bo2, (float*)d_out, N);
}